// Aggregation_2018634629403
// MI455X (gfx1250) — compile-verified
//
#include <hip/hip_runtime.h>

typedef __attribute__((ext_vector_type(2))) float v2f;
typedef __attribute__((ext_vector_type(8))) float v8f;

#define BZ 16
#define NN 2048
#define FF 128
#define NEGSLOPE 0.2f
#define ROWS (BZ * NN) // 32768
#define LPAD 132       // LDS row stride (floats): 528B = 33*16 -> 16B aligned rows, conflict-free

// Probe for the gfx1250 async memory->LDS path (ASYNCcnt-tracked).
#if defined(__has_builtin)
#if __has_builtin(__builtin_amdgcn_global_load_async_to_lds_b128) && \
    __has_builtin(__builtin_amdgcn_s_wait_asynccnt)
#define HAVE_ASYNC_LDS 1
#endif
#endif
#ifndef HAVE_ASYNC_LDS
#define HAVE_ASYNC_LDS 0
#endif

#if HAVE_ASYNC_LDS
// Builtin signature (from hipcc diagnostic): param0 = int4 addrspace(1)* (global src),
// param1 = int4 addrspace(3)* (LDS dst), then imm offset, imm cpol.
typedef int v4i __attribute__((vector_size(16)));
typedef __attribute__((address_space(1))) v4i as1_v4i;
typedef __attribute__((address_space(3))) v4i as3_v4i;
#endif

// ---------------- wave32 reductions ----------------
__device__ __forceinline__ float wave_sum(float v) {
#pragma unroll
    for (int m = 16; m > 0; m >>= 1) v += __shfl_xor(v, m, 32);
    return v;
}
__device__ __forceinline__ float wave_max(float v) {
#pragma unroll
    for (int m = 16; m > 0; m >>= 1) v = fmaxf(v, __shfl_xor(v, m, 32));
    return v;
}

// ---------------- K0: WT = W^T (so B-matrix lane pairs are contiguous) ----------------
__global__ __launch_bounds__(128) void k_wt(const float* __restrict__ W,
                                            float* __restrict__ WT) {
    const int g = blockIdx.x;  // 0..127 (output column of W)
    const int f = threadIdx.x; // 0..127
    WT[(size_t)g * FF + f] = W[(size_t)f * FF + g];
}

// ---------------- K1: H = X @ W via V_WMMA_F32_16X16X4_F32 ----------------
// grid.x = ROWS/16 (row tiles); 256 threads = 8 waves; wave w owns column tile w (8*16 = 128 = FF).
__global__ __launch_bounds__(256) void k_gemm(const float* __restrict__ X,
                                              const float* __restrict__ WT,
                                              float* __restrict__ H) {
    __shared__ float lds[16 * LPAD];
    const int row0 = blockIdx.x * 16;
    const int tid = threadIdx.x;

#if HAVE_ASYNC_LDS
    // 16 rows x 128 floats = 512 x 16B chunks; direct global->LDS DMA (ASYNCcnt).
    for (int t = tid; t < 512; t += 256) {
        const int r = t >> 5;       // row 0..15
        const int c = (t & 31) * 4; // float offset within row, 16B chunks
        __builtin_amdgcn_global_load_async_to_lds_b128(
            (as1_v4i*)(X + (size_t)(row0 + r) * FF + c),
            (as3_v4i*)(&lds[r * LPAD + c]), 0, 0);
    }
    __builtin_amdgcn_s_wait_asynccnt(0);
    __syncthreads();
#else
    for (int t = tid; t < 16 * 128; t += 256) {
        const int r = t >> 7, c = t & 127;
        lds[r * LPAD + c] = X[(size_t)(row0 + r) * FF + c];
    }
    __syncthreads();
#endif

    const int wave = tid >> 5;
    const int lane = tid & 31;
    const int m16 = lane & 15;
    const int hi = lane >> 4;       // 0: lanes 0-15, 1: lanes 16-31
    const int col0 = wave * 16;
    const float* wtrow = WT + (size_t)(col0 + m16) * FF; // B column (col0+m16), K-major

    v8f c = {};
#pragma unroll
    for (int k = 0; k < FF; k += 4) {
        const int ka = k + 2 * hi;  // A/B: VGPR0/1 hold K{0,1} (low lanes) / K{2,3} (high lanes)
        v2f a, b;
        a.x = lds[m16 * LPAD + ka];
        a.y = lds[m16 * LPAD + ka + 1];
        b = *(const v2f*)(wtrow + ka); // contiguous pair -> single b64 load
        c = __builtin_amdgcn_wmma_f32_16x16x4_f32(false, a, false, b, (short)0, c,
                                                  false, false);
    }
    // C/D layout: VGPR r -> M = r + 8*hi, N = m16
#pragma unroll
    for (int r = 0; r < 8; ++r)
        H[(size_t)(row0 + r + 8 * hi) * FF + col0 + m16] = c[r];
}

// ---------------- K2: s1 = H.b1, s2 = H.b2 (wave per row, float4 lanes) ----------------
__global__ __launch_bounds__(256) void k_proj(const float* __restrict__ H,
                                              const float* __restrict__ bvec,
                                              float* __restrict__ s1,
                                              float* __restrict__ s2) {
    const int row = blockIdx.x * 8 + (threadIdx.x >> 5);
    const int lane = threadIdx.x & 31;
    const float4 hv = ((const float4*)(H + (size_t)row * FF))[lane]; // 32*4 = 128 = FF
    const float4 x1 = ((const float4*)bvec)[lane];
    const float4 x2 = ((const float4*)(bvec + FF))[lane];
    float a1 = hv.x * x1.x + hv.y * x1.y + hv.z * x1.z + hv.w * x1.w;
    float a2 = hv.x * x2.x + hv.y * x2.y + hv.z * x2.z + hv.w * x2.w;
    a1 = wave_sum(a1);
    a2 = wave_sum(a2);
    if (lane == 0) { s1[row] = a1; s2[row] = a2; }
}

// ---------------- K3: per-batch max of s2 (wave per batch) ----------------
__global__ __launch_bounds__(512) void k_s2max(const float* __restrict__ s2,
                                               float* __restrict__ s2m) {
    const int b = threadIdx.x >> 5;
    const int lane = threadIdx.x & 31;
    float m = -3.0e38f;
    for (int j = lane; j < NN; j += 32) m = fmaxf(m, s2[b * NN + j]);
    m = wave_max(m);
    if (lane == 0) s2m[b] = m;
}

// ---------------- K4: per-row softmax stats (m_i, 1/Z_i); wave per row ----------------
// m_i = lrelu(s1_i + max_j s2_j) since lrelu is monotonic.
__global__ __launch_bounds__(256) void k_rows(const float* __restrict__ s1,
                                              const float* __restrict__ s2,
                                              const float* __restrict__ s2m,
                                              float* __restrict__ rowm,
                                              float* __restrict__ rinvZ) {
    const int row = blockIdx.x * 8 + (threadIdx.x >> 5);
    const int lane = threadIdx.x & 31;
    const int b = row >> 11; // /NN
    const float s1v = s1[row];
    float t = s1v + s2m[b];
    const float m = t > 0.f ? t : NEGSLOPE * t;
    const float* s2b = s2 + (size_t)b * NN;
    float acc = 0.f;
    for (int j = lane; j < NN; j += 32) {
        float x = s1v + s2b[j];
        x = x > 0.f ? x : NEGSLOPE * x;
        acc += __expf(x - m); // x <= m always -> no overflow
    }
    acc = wave_sum(acc);
    if (lane == 0) { rowm[row] = m; rinvZ[row] = 1.0f / acc; }
}

// ---------------- K5: colsum_j = sum_i attn[i,j] (wave per column) ----------------
__global__ __launch_bounds__(256) void k_cols(const float* __restrict__ s1,
                                              const float* __restrict__ s2,
                                              const float* __restrict__ rowm,
                                              const float* __restrict__ rinvZ,
                                              float* __restrict__ csum) {
    const int col = blockIdx.x * 8 + (threadIdx.x >> 5);
    const int lane = threadIdx.x & 31;
    const int b = col >> 11;
    const float s2v = s2[col];
    const float* s1b = s1 + (size_t)b * NN;
    const float* mb = rowm + (size_t)b * NN;
    const float* zb = rinvZ + (size_t)b * NN;
    float acc = 0.f;
    for (int i = lane; i < NN; i += 32) {
        float x = s1b[i] + s2v;
        x = x > 0.f ? x : NEGSLOPE * x;
        acc += __expf(x - mb[i]) * zb[i];
    }
    acc = wave_sum(acc);
    if (lane == 0) csum[col] = acc;
}

// ---------------- K6: out[b,f] = elu( sum_j colsum[b,j]*H[b,j,f] ) ----------------
__global__ __launch_bounds__(128) void k_out(const float* __restrict__ H,
                                             const float* __restrict__ csum,
                                             float* __restrict__ out) {
    const int b = blockIdx.x;
    const int f = threadIdx.x;
    const float* Hb = H + (size_t)b * NN * FF;
    const float* cb = csum + (size_t)b * NN;
    float acc = 0.f;
    for (int j = 0; j < NN; ++j)
        acc += cb[j] * Hb[(size_t)j * FF + f];
    out[b * FF + f] = acc > 0.f ? acc : (__expf(acc) - 1.0f);
}

extern "C" void kernel_launch(void* const* d_in, const int* in_sizes, int n_in,
                              void* d_out, int out_size, void* d_ws, size_t ws_size,
                              hipStream_t stream) {
    const float* X = (const float*)d_in[0];    // (B,N,F)
    const float* W = (const float*)d_in[1];    // (F,F)
    const float* bvec = (const float*)d_in[2]; // (2F,1)
    float* out = (float*)d_out;                // (B,F)

    // workspace (floats): H | s1 | s2 | rowm | rinvZ | csum | s2max | WT  (~17.5 MB)
    float* H = (float*)d_ws;
    float* s1 = H + (size_t)ROWS * FF;
    float* s2 = s1 + ROWS;
    float* rowm = s2 + ROWS;
    float* rinvZ = rowm + ROWS;
    float* csum = rinvZ + ROWS;
    float* s2m = csum + ROWS;
    float* WT = s2m + 64;

    k_wt<<<FF, 128, 0, stream>>>(W, WT);
    k_gemm<<<ROWS / 16, 256, 0, stream>>>(X, WT, H);
    k_proj<<<ROWS / 8, 256, 0, stream>>>(H, bvec, s1, s2);
    k_s2max<<<1, 512, 0, stream>>>(s2, s2m);
    k_rows<<<ROWS / 8, 256, 0, stream>>>(s1, s2, s2m, rowm, rinvZ);
    k_cols<<<ROWS / 8, 256, 0, stream>>>(s1, s2, rowm, rinvZ, csum);
    k_out<<<BZ, 128, 0, stream>>>(H, csum, out);
}